// GumbelVectorQuantizer_8933531976395
// MI455X (gfx1250) — compile-verified
//
#include <hip/hip_runtime.h>

typedef __attribute__((ext_vector_type(2))) float v2f;
typedef __attribute__((ext_vector_type(8))) float v8f;

#define CH      512
#define NV      320          // codewords per group
#define NG      2
#define ND      128          // codevector dim per group
#define NCOL    640          // NG*NV
#define MTILE   32
#define KSTEPS  128          // CH/4
#define NTW     5            // n-tiles (of 16 cols) per wave; 8 waves * 5 * 16 = 640
#define A_STRIDE 516         // 512 + 4 words: wave's A-frag ds_read_b64 covers all 64 banks once
#define V_STRIDE 644

// ---------------------------------------------------------------------------
// Kernel 0: pack W (row-major [512][640]) into WMMA B-fragment lane order:
//   Wf[((ntile*KSTEPS + kstep)*32 + lane)] = (W[k][n], W[k+1][n])
//   n = ntile*16 + (lane&15),  k = kstep*4 + (lane>>4)*2
// (B 4x16 layout: K=0,1 in vgpr0,1 lanes 0-15; K=2,3 in lanes 16-31)
// ---------------------------------------------------------------------------
__global__ void gvq_pack_w(const float* __restrict__ W, v2f* __restrict__ Wf) {
    int tid = blockIdx.x * blockDim.x + threadIdx.x;
    if (tid >= (NCOL / 16) * KSTEPS * 32) return;
    int lane = tid & 31;
    int ks   = (tid >> 5) & (KSTEPS - 1);
    int nt   = tid >> 12;                 // 4096 entries per n-tile
    int n = nt * 16 + (lane & 15);
    int k = ks * 4 + ((lane >> 4) << 1);
    v2f v;
    v.x = W[(size_t)k * NCOL + n];
    v.y = W[(size_t)(k + 1) * NCOL + n];
    Wf[tid] = v;
}

// ---------------------------------------------------------------------------
// Kernel 1: fused  (x @ W + b) -> +gumbel -> argmax per group -> gather rows
// One block = 32 time-positions x all 640 logit columns. 8 waves.
// ---------------------------------------------------------------------------
__global__ __launch_bounds__(256)
void gvq_main(const float* __restrict__ x,
              const v2f*  __restrict__ Wf,
              const float* __restrict__ bias,
              const float* __restrict__ codevectors,
              const float* __restrict__ unoise,
              float* __restrict__ out)
{
    __shared__ float smem[MTILE * V_STRIDE];   // A-tile region reused as Vals region
    __shared__ int   sidx[MTILE * NG];

    const int tid   = threadIdx.x;
    const int lane  = tid & 31;
    const int wave  = tid >> 5;
    const int lm    = lane & 15;
    const int hlf   = lane >> 4;               // 0: K=0..1 / rows M, 1: K=2..3 / rows M+8
    const int m_base = blockIdx.x * MTILE;

    // ---- phase 0: stage x tile (32 x 512 f32) into LDS, padded stride ----
    for (int i = tid; i < MTILE * (CH / 4); i += 256) {
        int row = i >> 7;                      // 128 float4 per row
        int k4  = i & 127;
        float4 v = ((const float4*)x)[(size_t)(m_base + row) * (CH / 4) + k4];
        *(float4*)&smem[row * A_STRIDE + k4 * 4] = v;
    }
    __syncthreads();

    // ---- phase 1: GEMM, fp32 WMMA 16x16x4, acc[2 row-subtiles][5 n-tiles] ----
    v8f acc[2][NTW];
    #pragma unroll
    for (int ms = 0; ms < 2; ++ms)
        #pragma unroll
        for (int nt = 0; nt < NTW; ++nt)
            acc[ms][nt] = 0.0f;

    const v2f* wfw = Wf + (size_t)wave * NTW * KSTEPS * 32;

    for (int ks = 0; ks < KSTEPS; ++ks) {
        const int k = ks * 4 + hlf * 2;
        v2f a0 = *(const v2f*)&smem[(lm)      * A_STRIDE + k];
        v2f a1 = *(const v2f*)&smem[(16 + lm) * A_STRIDE + k];
        v2f b[NTW];
        #pragma unroll
        for (int nt = 0; nt < NTW; ++nt)
            b[nt] = wfw[(size_t)(nt * KSTEPS + ks) * 32 + lane];
        #pragma unroll
        for (int nt = 0; nt < NTW; ++nt) {
            acc[0][nt] = __builtin_amdgcn_wmma_f32_16x16x4_f32(
                false, a0, false, b[nt], (short)0, acc[0][nt], false, false);
            acc[1][nt] = __builtin_amdgcn_wmma_f32_16x16x4_f32(
                false, a1, false, b[nt], (short)0, acc[1][nt], false, false);
        }
    }
    __syncthreads();   // everyone done reading the A region of LDS

    // ---- phase 2: logits + bias + gumbel noise -> Vals in LDS ----
    // C layout: vgpr r -> row (r + hlf*8), col = lane&15 within 16-col tile
    #pragma unroll
    for (int ms = 0; ms < 2; ++ms) {
        #pragma unroll
        for (int nt = 0; nt < NTW; ++nt) {
            const int col  = wave * (NTW * 16) + nt * 16 + lm;
            const float bc = bias[col];
            #pragma unroll
            for (int r = 0; r < 8; ++r) {
                const int m_local = ms * 16 + r + hlf * 8;
                float u   = unoise[(size_t)(m_base + m_local) * NCOL + col];
                float val = acc[ms][nt][r] + bc - __logf(-__logf(u));
                smem[m_local * V_STRIDE + col] = val;
            }
        }
    }
    __syncthreads();

    // ---- phase 3: argmax over 320 per (row, group); first-max wins ----
    if (tid < MTILE * NG) {
        const int m_local = tid >> 1;
        const int g       = tid & 1;
        const float* v = &smem[m_local * V_STRIDE + g * NV];
        float best = v[0];
        int   bi   = 0;
        for (int i = 1; i < NV; ++i) {
            float t = v[i];
            if (t > best) { best = t; bi = i; }
        }
        sidx[tid] = bi;
    }
    __syncthreads();

    // ---- phase 4: gather selected codevector rows -> out [row][g*128+d] ----
    for (int i = tid; i < MTILE * NG * (ND / 4); i += 256) {
        int task    = i >> 5;                  // 32 float4 per task
        int q       = i & 31;
        int m_local = task >> 1;
        int g       = task & 1;
        int idx     = sidx[task];
        float4 cv = ((const float4*)codevectors)[(size_t)(g * NV + idx) * (ND / 4) + q];
        ((float4*)out)[(size_t)(m_base + m_local) * (NG * ND / 4) + g * (ND / 4) + q] = cv;
    }
}

// ---------------------------------------------------------------------------
extern "C" void kernel_launch(void* const* d_in, const int* in_sizes, int n_in,
                              void* d_out, int out_size, void* d_ws, size_t ws_size,
                              hipStream_t stream) {
    const float* x    = (const float*)d_in[0];   // [B,T,512]
    const float* W    = (const float*)d_in[1];   // [512,640]
    const float* bias = (const float*)d_in[2];   // [640]
    const float* cv   = (const float*)d_in[3];   // [2,320,128]
    const float* u    = (const float*)d_in[4];   // [B,T,2,320]
    float* out        = (float*)d_out;           // [B,T,256]

    const int M = in_sizes[0] / CH;              // B*T = 32768

    v2f* Wf = (v2f*)d_ws;                        // 40*128*32 float2 = 1.25 MiB

    const int pack_elems = (NCOL / 16) * KSTEPS * 32;
    gvq_pack_w<<<(pack_elems + 255) / 256, 256, 0, stream>>>(W, Wf);
    gvq_main<<<M / MTILE, 256, 0, stream>>>(x, Wf, bias, cv, u, out);
}